// LIF_23304492548820
// MI455X (gfx1250) — compile-verified
//
#include <hip/hip_runtime.h>

// LIF scan over T=32 for x[T, B*N] fp32. Pure streaming kernel (~268 MB traffic,
// ~11.5us at 23.3 TB/s). CDNA5 path: async global->LDS b128 loads with a 4-deep
// per-wave pipeline (s_wait_asynccnt), nontemporal b128 stores. No barriers:
// each lane reads back only the LDS slot its own wave's async load filled.

typedef __attribute__((ext_vector_type(4))) float f32x4;

constexpr int T     = 32;   // timesteps (scan length)
constexpr int BLOCK = 256;  // 8 waves (wave32)
constexpr int PIPE  = 4;    // async pipeline depth (ring of 4 LDS buffers)

__device__ __forceinline__ void async_copy_f4(unsigned lds_off, const f32x4* gsrc) {
  // GV mode: vdst = per-lane LDS byte offset, vaddr = 64-bit global address.
  asm volatile("global_load_async_to_lds_b128 %0, %1, off"
               :: "v"(lds_off), "v"(gsrc)
               : "memory");
}

template <int N>
__device__ __forceinline__ void wait_async() {
  asm volatile("s_wait_asynccnt %0" :: "n"(N) : "memory");
}

__device__ __forceinline__ void wait_ds0() {
  asm volatile("s_wait_dscnt 0x0" ::: "memory");
}

__global__ __launch_bounds__(BLOCK) void lif_scan_kernel(
    const float* __restrict__ x, float* __restrict__ out, int bn4) {
  __shared__ f32x4 buf[PIPE][BLOCK];

  const int gid = blockIdx.x * BLOCK + threadIdx.x;
  if (gid >= bn4) return;

  const f32x4* __restrict__ xv = reinterpret_cast<const f32x4*>(x);
  f32x4* __restrict__ ov       = reinterpret_cast<f32x4*>(out);

  // Per-lane LDS byte offsets for the 4 ring slots. Low 32 bits of a flat
  // pointer into LDS are the workgroup-relative LDS byte address.
  unsigned lds_off[PIPE];
#pragma unroll
  for (int p = 0; p < PIPE; ++p)
    lds_off[p] = (unsigned)(size_t)(&buf[p][threadIdx.x]);

  // Prologue: put the first PIPE timesteps in flight.
#pragma unroll
  for (int k = 0; k < PIPE; ++k)
    async_copy_f4(lds_off[k], xv + (size_t)k * bn4 + gid);

  f32x4 m = {0.0f, 0.0f, 0.0f, 0.0f};

#pragma unroll
  for (int t = 0; t < T; ++t) {
    // Oldest in-flight load (timestep t) must be complete.
    if (t < T - 3)      wait_async<PIPE - 1>();
    else if (t == T - 3) wait_async<2>();
    else if (t == T - 2) wait_async<1>();
    else                 wait_async<0>();

    // Consume our own lane's slot from LDS into registers.
    f32x4 v = buf[t & (PIPE - 1)][threadIdx.x];
    // Make sure the ds_load has drained before the async write below
    // re-targets this same ring slot (WAR on LDS).
    wait_ds0();

    // Keep the pipeline full: issue timestep t+PIPE into the freed slot.
    if (t + PIPE < T)
      async_copy_f4(lds_off[t & (PIPE - 1)], xv + (size_t)(t + PIPE) * bn4 + gid);

    // LIF update. m*0.25 is exact in fp32, so fma == mul-then-add exactly
    // (bit-identical threshold behaviour vs. the reference).
    f32x4 o;
    m.x = __builtin_fmaf(m.x, 0.25f, v.x);
    m.y = __builtin_fmaf(m.y, 0.25f, v.y);
    m.z = __builtin_fmaf(m.z, 0.25f, v.z);
    m.w = __builtin_fmaf(m.w, 0.25f, v.w);
    o.x = (m.x > 1.0f) ? 1.0f : 0.0f;
    o.y = (m.y > 1.0f) ? 1.0f : 0.0f;
    o.z = (m.z > 1.0f) ? 1.0f : 0.0f;
    o.w = (m.w > 1.0f) ? 1.0f : 0.0f;
    m.x = (m.x > 1.0f) ? 0.0f : m.x;  // hard reset
    m.y = (m.y > 1.0f) ? 0.0f : m.y;
    m.z = (m.z > 1.0f) ? 0.0f : m.z;
    m.w = (m.w > 1.0f) ? 0.0f : m.w;

    // Output is write-once / never re-read: nontemporal b128 store.
    __builtin_nontemporal_store(o, ov + (size_t)t * bn4 + gid);
  }
}

extern "C" void kernel_launch(void* const* d_in, const int* in_sizes, int n_in,
                              void* d_out, int out_size, void* d_ws, size_t ws_size,
                              hipStream_t stream) {
  const float* x = (const float*)d_in[0];
  float* out     = (float*)d_out;

  const int total = in_sizes[0];   // T * B * N = 32 * 64 * 16384
  const int bn    = total / T;     // B * N  = 1,048,576
  const int bn4   = bn / 4;        // float4 lanes = 262,144
  const int grid  = (bn4 + BLOCK - 1) / BLOCK;  // 1024 blocks

  lif_scan_kernel<<<grid, BLOCK, 0, stream>>>(x, out, bn4);
}